// BaselineGINELayer_10866267259113
// MI455X (gfx1250) — compile-verified
//
#include <hip/hip_runtime.h>

typedef __attribute__((ext_vector_type(2))) float v2f;
typedef __attribute__((ext_vector_type(8))) float v8f;

#define D 128
#define LDSS 132   // padded LDS row stride (floats): conflict-free b64 reads, 8B aligned

// ---------------- kernel 1: z = (1+eps) * x ----------------
__global__ void gine_init_z(const float* __restrict__ x, const float* __restrict__ eps,
                            float* __restrict__ z, int total4) {
    int i = blockIdx.x * blockDim.x + threadIdx.x;
    if (i >= total4) return;
    float s = 1.0f + eps[0];
    float4 v = ((const float4*)x)[i];
    v.x *= s; v.y *= s; v.z *= s; v.w *= s;
    ((float4*)z)[i] = v;
}

// ---------------- kernel 2: scatter-add relu(x[src]+e) into z[dst] ----------------
__global__ void gine_edge_scatter(const float* __restrict__ x, const int* __restrict__ ei,
                                  const float* __restrict__ ea, float* __restrict__ z, int E) {
    long long g = (long long)blockIdx.x * blockDim.x + threadIdx.x;
    long long e = g >> 5;          // edge id
    int c = (int)(g & 31);         // float4 chunk 0..31
    if (e >= E) return;
    int src = ei[e];
    int dst = ei[(long long)E + e];
    float4 xv = ((const float4*)(x + (long long)src * D))[c];
    float4 ev = ((const float4*)(ea + e * (long long)D))[c];
    float4 m;
    m.x = fmaxf(xv.x + ev.x, 0.0f);
    m.y = fmaxf(xv.y + ev.y, 0.0f);
    m.z = fmaxf(xv.z + ev.z, 0.0f);
    m.w = fmaxf(xv.w + ev.w, 0.0f);
    float* o = z + (long long)dst * D + c * 4;
    atomicAdd(o + 0, m.x);
    atomicAdd(o + 1, m.y);
    atomicAdd(o + 2, m.z);
    atomicAdd(o + 3, m.w);
}

// ---------------- kernel 3/4: out = relu(bn((resid?) + A @ W^T + bias)) ----------------
// 256 threads = 8 waves; block computes 128 rows x 128 cols.
// Wave w owns rows [w*16, w*16+16); 8 accumulator tiles of 16x16 (v8f each).
__global__ __launch_bounds__(256)
void gine_gemm_bn_relu(const float* __restrict__ A,     // [n,128] activations
                       const float* __restrict__ W,     // [128,128] row-major [o][k]
                       const float* __restrict__ bias,
                       const float* __restrict__ gamma,
                       const float* __restrict__ beta,
                       const float* __restrict__ rmean,
                       const float* __restrict__ rvar,
                       const float* __restrict__ resid, // nullptr for layer 1
                       float* __restrict__ out,         // [n,128] (may alias A; block-row safe)
                       int n_rows) {
    __shared__ float Asm[D * LDSS];
    __shared__ float Bsm[D * LDSS];
    __shared__ float s_sc[D];
    __shared__ float s_sh[D];

    const int tid  = threadIdx.x;
    const int wave = tid >> 5;
    const int lane = tid & 31;
    const int lo   = lane & 15;
    const int hi   = lane >> 4;
    const long long rowBase = (long long)blockIdx.x * 128;

    // stage W (B-matrix source): 8192 float2
    for (int i = tid; i < D * (D / 2); i += 256) {
        int r = i >> 6, j = i & 63;
        float2 w = ((const float2*)W)[i];
        *(float2*)&Bsm[r * LDSS + j * 2] = w;
    }
    // stage 128 activation rows (zero-pad past n_rows)
    for (int i = tid; i < D * (D / 2); i += 256) {
        int r = i >> 6, j = i & 63;
        long long row = rowBase + r;
        float2 v = make_float2(0.0f, 0.0f);
        if (row < n_rows) v = ((const float2*)(A + row * (long long)D))[j];
        *(float2*)&Asm[r * LDSS + j * 2] = v;
    }
    // fold BN: y = relu((v + bias - rmean)*s + beta) = relu(v*s + c)
    if (tid < D) {
        float s = gamma[tid] * rsqrtf(rvar[tid] + 1e-5f);
        s_sc[tid] = s;
        s_sh[tid] = (bias[tid] - rmean[tid]) * s + beta[tid];
    }
    __syncthreads();

    v8f acc[8];
#pragma unroll
    for (int t = 0; t < 8; ++t) acc[t] = (v8f)0.0f;

    const int mrow = wave * 16 + lo;  // A row this lane supplies
#pragma unroll 4
    for (int k = 0; k < D; k += 4) {
        // A frag 16x4: lanes 0-15 -> K=k,k+1 ; lanes 16-31 -> K=k+2,k+3
        v2f a = *(const v2f*)&Asm[mrow * LDSS + k + 2 * hi];
#pragma unroll
        for (int t = 0; t < 8; ++t) {
            // B frag 4x16: B[k][n] = W[n][k]; column n = t*16+lo
            v2f b = *(const v2f*)&Bsm[(t * 16 + lo) * LDSS + k + 2 * hi];
            acc[t] = __builtin_amdgcn_wmma_f32_16x16x4_f32(
                false, a, false, b, (short)0, acc[t], false, false);
        }
    }

    // epilogue: C/D layout — VGPR i: lanes0-15 M=i, lanes16-31 M=8+i; N = lane&15
#pragma unroll
    for (int t = 0; t < 8; ++t) {
        int n = t * 16 + lo;
        float s = s_sc[n], c = s_sh[n];
#pragma unroll
        for (int i = 0; i < 8; ++i) {
            long long row = rowBase + wave * 16 + i + 8 * hi;
            if (row < n_rows) {
                float v = acc[t][i];
                if (resid) v += resid[row * (long long)D + n];
                v = fmaxf(v * s + c, 0.0f);
                out[row * (long long)D + n] = v;
            }
        }
    }
}

extern "C" void kernel_launch(void* const* d_in, const int* in_sizes, int n_in,
                              void* d_out, int out_size, void* d_ws, size_t ws_size,
                              hipStream_t stream) {
    const float* x   = (const float*)d_in[0];
    const int*   ei  = (const int*)d_in[1];
    const float* ea  = (const float*)d_in[2];
    const float* eps = (const float*)d_in[3];
    const float* W1  = (const float*)d_in[4];
    const float* b1  = (const float*)d_in[5];
    const float* g1  = (const float*)d_in[6];
    const float* bt1 = (const float*)d_in[7];
    const float* rm1 = (const float*)d_in[8];
    const float* rv1 = (const float*)d_in[9];
    const float* W2  = (const float*)d_in[10];
    const float* b2  = (const float*)d_in[11];
    const float* g2  = (const float*)d_in[12];
    const float* bt2 = (const float*)d_in[13];
    const float* rm2 = (const float*)d_in[14];
    const float* rv2 = (const float*)d_in[15];

    const int n = in_sizes[0] / D;   // 100000
    const int E = in_sizes[2] / D;   // 600000

    float* z   = (float*)d_ws;       // [n,128] aggregation accumulator
    float* h1  = (float*)d_out;      // layer-1 activations live in d_out
    float* out = (float*)d_out;

    // 1) z = (1+eps)*x
    int total4 = n * (D / 4);
    gine_init_z<<<(total4 + 255) / 256, 256, 0, stream>>>(x, eps, z, total4);

    // 2) z += segment_sum(relu(x[src]+e), dst)
    long long ethreads = (long long)E * (D / 4);
    int eblocks = (int)((ethreads + 255) / 256);
    gine_edge_scatter<<<eblocks, 256, 0, stream>>>(x, ei, ea, z, E);

    // 3) h1 = relu(bn1(z @ W1^T + b1))
    int gblocks = (n + 127) / 128;
    gine_gemm_bn_relu<<<gblocks, 256, 0, stream>>>(z, W1, b1, g1, bt1, rm1, rv1,
                                                   nullptr, h1, n);

    // 4) out = relu(bn2(x + h1 @ W2^T + b2))   (in-place over d_out; block-row safe)
    gine_gemm_bn_relu<<<gblocks, 256, 0, stream>>>(h1, W2, b2, g2, bt2, rm2, rv2,
                                                   x, out, n);
}